// Pointnet_Tracking_47639777247838
// MI455X (gfx1250) — compile-verified
//
#include <hip/hip_runtime.h>
#include <math.h>

#define BATCH 32

typedef float v2f __attribute__((ext_vector_type(2)));
typedef float v8f __attribute__((ext_vector_type(8)));

// CDNA5 fp32 WMMA: D(16x16) = A(16x4) * B(4x16) + C, wave32.
//   A: lane l, vgpr v -> row = l%16, k = 2*(l/16) + v
//   B: lane l, vgpr v -> col = l%16, k = v + 2*(l/16)
//   C/D: lane l, vgpr r -> row = r + 8*(l/16), col = l%16
__device__ __forceinline__ v8f wmma_f32(v2f a, v2f b, v8f c) {
  return __builtin_amdgcn_wmma_f32_16x16x4_f32(false, a, false, b, (short)0, c,
                                               false, false);
}

// ---------------------------------------------------------------------------
// Pad weight matrix W (Cout x K) -> Wp (Cout x Kp) with zeros (K padded to 4).
__global__ void k_padw(const float* __restrict__ W, float* __restrict__ Wp,
                       int Cout, int K, int Kp) {
  int t = blockIdx.x * blockDim.x + threadIdx.x;
  if (t >= Cout * Kp) return;
  int r = t / Kp, c = t - r * Kp;
  Wp[t] = (c < K) ? W[(long)r * K + c] : 0.f;
}

// Split pc (B,N,6) -> xyz (B,N,3), feat (B,N,3)
__global__ void k_split(const float* __restrict__ pc, float* __restrict__ xyz,
                        float* __restrict__ feat, int N) {
  int t = blockIdx.x * blockDim.x + threadIdx.x;
  int total = BATCH * N * 3;
  if (t >= total) return;
  int p = t / 3, c = t - p * 3;
  xyz[t]  = pc[(long)p * 6 + c];
  feat[t] = pc[(long)p * 6 + 3 + c];
}

// ---------------------------------------------------------------------------
// Farthest point sampling. One block per batch. Matches jax.lax.scan:
// emits current `far` then updates (argmax picks FIRST max index).
__global__ void k_fps(const float* __restrict__ xyz, int* __restrict__ outIdx,
                      int N, int npoint) {
  __shared__ float dist[1024];
  __shared__ float sv[256];
  __shared__ int   si[256];
  int b = blockIdx.x, tid = threadIdx.x;
  const float* P = xyz + (long)b * N * 3;
  for (int j = tid; j < N; j += blockDim.x) dist[j] = 1e10f;
  __syncthreads();
  int far = 0;
  for (int i = 0; i < npoint; i++) {
    if (tid == 0) outIdx[b * npoint + i] = far;
    float cx = P[far * 3 + 0], cy = P[far * 3 + 1], cz = P[far * 3 + 2];
    float bv = -1.f; int bi = 0;
    for (int j = tid; j < N; j += blockDim.x) {
      float dx = P[j * 3 + 0] - cx, dy = P[j * 3 + 1] - cy, dz = P[j * 3 + 2] - cz;
      float d = dx * dx + dy * dy + dz * dz;
      float dm = fminf(dist[j], d);
      dist[j] = dm;
      if (dm > bv) { bv = dm; bi = j; }   // strict > keeps smallest index on ties
    }
    sv[tid] = bv; si[tid] = bi;
    __syncthreads();
    for (int o = blockDim.x >> 1; o > 0; o >>= 1) {
      if (tid < o) {
        float v2 = sv[tid + o]; int i2 = si[tid + o];
        if (v2 > sv[tid] || (v2 == sv[tid] && i2 < si[tid])) { sv[tid] = v2; si[tid] = i2; }
      }
      __syncthreads();
    }
    far = si[0];
    __syncthreads();
  }
}

// new_xyz[b,s,:] = xyz[b, idx[b,s], :]
__global__ void k_gather3(const float* __restrict__ xyz, const int* __restrict__ idx,
                          float* __restrict__ out, int N, int S) {
  int t = blockIdx.x * blockDim.x + threadIdx.x;
  int total = BATCH * S * 3;
  if (t >= total) return;
  int c = t % 3, p = t / 3, b = p / S;
  out[t] = xyz[((long)b * N + idx[p]) * 3 + c];
}

// Ball query: ns smallest in-range indices (ascending scan == sorted),
// padded with the first in-range index (0 if none) — matches reference.
__global__ void k_ballq(const float* __restrict__ xyz, const float* __restrict__ nx,
                        int* __restrict__ idx, int N, int S, int ns, float r2) {
  int p = blockIdx.x * blockDim.x + threadIdx.x;
  if (p >= BATCH * S) return;
  int b = p / S;
  const float* P = xyz + (long)b * N * 3;
  float cx = nx[p * 3 + 0], cy = nx[p * 3 + 1], cz = nx[p * 3 + 2];
  int* o = idx + (long)p * ns;
  int cnt = 0, first = -1;
  for (int j = 0; j < N; j++) {
    float dx = P[j * 3 + 0] - cx, dy = P[j * 3 + 1] - cy, dz = P[j * 3 + 2] - cz;
    if (dx * dx + dy * dy + dz * dz <= r2) {
      if (first < 0) first = j;
      o[cnt++] = j;
      if (cnt == ns) break;
    }
  }
  if (first < 0) first = 0;
  for (int k = cnt; k < ns; k++) o[k] = first;
}

// ---------------------------------------------------------------------------
// One MLP layer inside LDS: nxt = relu(cur * W^T + bias).
// W must be pre-padded: row stride Kp (multiple of 4), Co multiple of 64 is
// required — each wave does groups of exactly 4 N-tiles (Co % 64 == 0
// here for all uses). No guards -> unconditional b64 loads, branchless.
__device__ __forceinline__ void mlp_layer(
    const float* cur, float* nxt,
    const float* __restrict__ W, const float* __restrict__ Bv,
    int Kp, int Co, int nrows, int bufstride,
    int wave, int nw, int half, int lr) {
  int tilesG = Co >> 6;                 // groups of 4 N-tiles (Co % 64 == 0)
  int tiles = (nrows >> 4) * tilesG;
  for (int t = wave; t < tiles; t += nw) {
    int mt = t / tilesG, ng = t - mt * tilesG;
    int n0 = ng * 64 + lr;
    const float* arow = cur + (mt * 16 + lr) * bufstride;
    v8f acc[4];
#pragma unroll
    for (int i = 0; i < 4; i++) acc[i] = (v8f){0.f,0.f,0.f,0.f,0.f,0.f,0.f,0.f};
    for (int k0 = 0; k0 < Kp; k0 += 4) {
      int ka = k0 + 2 * half;
      v2f a = *(const v2f*)(arow + ka);               // ds_load_b64
      v2f b0 = *(const v2f*)(W + (long)(n0     ) * Kp + ka);
      v2f b1 = *(const v2f*)(W + (long)(n0 + 16) * Kp + ka);
      v2f b2 = *(const v2f*)(W + (long)(n0 + 32) * Kp + ka);
      v2f b3 = *(const v2f*)(W + (long)(n0 + 48) * Kp + ka);
      acc[0] = wmma_f32(a, b0, acc[0]);
      acc[1] = wmma_f32(a, b1, acc[1]);
      acc[2] = wmma_f32(a, b2, acc[2]);
      acc[3] = wmma_f32(a, b3, acc[3]);
    }
#pragma unroll
    for (int i = 0; i < 4; i++) {
      float bv = Bv[n0 + 16 * i];
      float* orow = nxt + n0 + 16 * i;
#pragma unroll
      for (int r = 0; r < 8; r++) {
        float v = fmaxf(acc[i][r] + bv, 0.f);         // all layers ReLU
        orow[(mt * 16 + r + 8 * half) * bufstride] = v;
      }
    }
  }
}

// Fused rows -> 3-layer MLP(relu) -> max kernel. One workgroup = one (b, s).
// mode 0: SA grouping rows = [xyz[j]-new_xyz, feat[j]]   (nrows = nsample)
// mode 1: xcorr fusion rows = [cos(b,r,s), t_xyz(b,r), t_feat(b,r)] (nrows = n1)
__global__ void k_rows_mlp_max(
    int mode,
    const float* __restrict__ xyz, const float* __restrict__ feat,
    const float* __restrict__ newxyz, const int* __restrict__ idx,
    const float* __restrict__ cosbuf,
    int S, int Npts, int Cf, int nrows,
    int c0p, int c1, int c2, int c3,
    const float* __restrict__ w0p, const float* __restrict__ b0,
    const float* __restrict__ w1,  const float* __restrict__ b1,
    const float* __restrict__ w2,  const float* __restrict__ b2,
    int bufstride, int bufElems, float* __restrict__ out) {
  extern __shared__ float lds[];
  float* buf0 = lds;
  float* buf1 = lds + bufElems;
  int s = blockIdx.x, b = blockIdx.y;

  // Stage input rows (zero-padded to c0p).
  for (int t = threadIdx.x; t < nrows * c0p; t += blockDim.x) {
    int r = t / c0p, c = t - r * c0p;
    float v = 0.f;
    if (mode == 0) {
      int j = idx[((long)b * S + s) * nrows + r];
      if (c < 3)            v = xyz[((long)b * Npts + j) * 3 + c] - newxyz[((long)b * S + s) * 3 + c];
      else if (c < 3 + Cf)  v = feat[((long)b * Npts + j) * Cf + (c - 3)];
    } else {
      if (c == 0)      v = cosbuf[((long)b * nrows + r) * S + s];
      else if (c < 4)  v = newxyz[((long)b * nrows + r) * 3 + (c - 1)];
      else             v = feat[((long)b * nrows + r) * Cf + (c - 4)];
    }
    buf0[r * bufstride + c] = v;
  }
  __syncthreads();

  int wave = threadIdx.x >> 5, nw = blockDim.x >> 5;
  int lane = threadIdx.x & 31, half = lane >> 4, lr = lane & 15;

  mlp_layer(buf0, buf1, w0p, b0, c0p, c1, nrows, bufstride, wave, nw, half, lr);
  __syncthreads();
  mlp_layer(buf1, buf0, w1,  b1, c1,  c2, nrows, bufstride, wave, nw, half, lr);
  __syncthreads();
  mlp_layer(buf0, buf1, w2,  b2, c2,  c3, nrows, bufstride, wave, nw, half, lr);
  __syncthreads();

  // Max over rows -> out[b,s,0:c3]
  for (int ch = threadIdx.x; ch < c3; ch += blockDim.x) {
    float m = buf1[ch];
    for (int r = 1; r < nrows; r++) m = fmaxf(m, buf1[r * bufstride + ch]);
    out[((long)b * S + s) * c3 + ch] = m;
  }
}

// ---------------------------------------------------------------------------
// Fast batched GEMM: Y[p,n] = act(sum_k X[p,k]*W[n,k] + bias[n]).
// Requires M%16==0, N%64==0, K%4==0. One wave = 1 M-tile x 4 N-tiles.
__global__ void k_gemm_fast(const float* __restrict__ X, const float* __restrict__ W,
                            const float* __restrict__ bias, float* __restrict__ Y,
                            int M, int N, int K, long sX, long sW, long sY, int relu) {
  int batch = blockIdx.y;
  const float* Xb = X + (long)batch * sX;
  const float* Wb = W + (long)batch * sW;
  float* Yb = Y + (long)batch * sY;
  int tilesG = N >> 6;
  int tile = blockIdx.x * (blockDim.x >> 5) + (threadIdx.x >> 5);
  if (tile >= (M >> 4) * tilesG) return;
  int mt = tile / tilesG, ng = tile - mt * tilesG;
  int lane = threadIdx.x & 31, half = lane >> 4, lr = lane & 15;
  int n0 = ng * 64 + lr;
  const float* arow = Xb + (long)(mt * 16 + lr) * K;
  v8f acc[4];
#pragma unroll
  for (int i = 0; i < 4; i++) acc[i] = (v8f){0.f,0.f,0.f,0.f,0.f,0.f,0.f,0.f};
  for (int k0 = 0; k0 < K; k0 += 4) {
    int ka = k0 + 2 * half;
    v2f a  = *(const v2f*)(arow + ka);
    v2f b0 = *(const v2f*)(Wb + (long)(n0     ) * K + ka);
    v2f b1 = *(const v2f*)(Wb + (long)(n0 + 16) * K + ka);
    v2f b2 = *(const v2f*)(Wb + (long)(n0 + 32) * K + ka);
    v2f b3 = *(const v2f*)(Wb + (long)(n0 + 48) * K + ka);
    acc[0] = wmma_f32(a, b0, acc[0]);
    acc[1] = wmma_f32(a, b1, acc[1]);
    acc[2] = wmma_f32(a, b2, acc[2]);
    acc[3] = wmma_f32(a, b3, acc[3]);
  }
#pragma unroll
  for (int i = 0; i < 4; i++) {
    int n = n0 + 16 * i;
    float bv = bias ? bias[n] : 0.f;   // uniform branch
#pragma unroll
    for (int r = 0; r < 8; r++) {
      float v = acc[i][r] + bv;
      if (relu) v = fmaxf(v, 0.f);
      Yb[(long)(mt * 16 + r + 8 * half) * N + n] = v;
    }
  }
}

// Edge GEMM for ragged N (cla N=1, prop N=5, vote N=259). M%16==0, K%4==0.
// Branchless: clamp n for loads (extra columns compute garbage that is never
// stored), guard only the store.
__global__ void k_gemm_edge(const float* __restrict__ X, const float* __restrict__ W,
                            const float* __restrict__ bias, float* __restrict__ Y,
                            int M, int N, int K) {
  int tilesN = (N + 15) >> 4;
  int tile = blockIdx.x * (blockDim.x >> 5) + (threadIdx.x >> 5);
  if (tile >= (M >> 4) * tilesN) return;
  int mt = tile / tilesN, nt = tile - mt * tilesN;
  int lane = threadIdx.x & 31, half = lane >> 4, lr = lane & 15;
  int n = nt * 16 + lr;
  int nc = n < N ? n : N - 1;
  const float* arow = X + (long)(mt * 16 + lr) * K;
  const float* wrow = W + (long)nc * K;
  v8f acc = {0.f,0.f,0.f,0.f,0.f,0.f,0.f,0.f};
  for (int k0 = 0; k0 < K; k0 += 4) {
    int ka = k0 + 2 * half;
    v2f a = *(const v2f*)(arow + ka);
    v2f b = *(const v2f*)(wrow + ka);
    acc = wmma_f32(a, b, acc);
  }
  float bv = bias[nc];
  if (n < N) {
#pragma unroll
    for (int r = 0; r < 8; r++)
      Y[(long)(mt * 16 + r + 8 * half) * N + n] = acc[r] + bv;   // no relu uses
  }
}

// ---------------------------------------------------------------------------
// Channel-wise L2 normalize per position: y = x / max(||x||, 1e-8)
__global__ void k_norm(const float* __restrict__ x, float* __restrict__ y,
                       int M, int C) {
  int p = blockIdx.x * blockDim.x + threadIdx.x;
  if (p >= M) return;
  const float* r = x + (long)p * C;
  float s = 0.f;
  for (int c = 0; c < C; c++) s += r[c] * r[c];
  float inv = 1.f / fmaxf(sqrtf(s), 1e-8f);
  float* o = y + (long)p * C;
  for (int c = 0; c < C; c++) o[c] = r[c] * inv;
}

// estimation_cla -> d_out[0:4096]; score = sigmoid(cla)
__global__ void k_cla_sig(const float* __restrict__ cla, float* __restrict__ out0,
                          float* __restrict__ score) {
  int p = blockIdx.x * blockDim.x + threadIdx.x;
  if (p >= BATCH * 128) return;
  float v = cla[p];
  out0[p] = v;
  score[p] = 1.f / (1.f + expf(-v));
}

// fxf[p, 0:260] = [search_xyz(3), fusion_feature(256), 0-pad(1)]  (stride 260)
__global__ void k_build_fxf(const float* __restrict__ sxyz, const float* __restrict__ ff,
                            float* __restrict__ fxf) {
  int t = blockIdx.x * blockDim.x + threadIdx.x;
  int total = BATCH * 128 * 260;
  if (t >= total) return;
  int p = t / 260, c = t - p * 260;
  float v = 0.f;
  if (c < 3)        v = sxyz[p * 3 + c];
  else if (c < 259) v = ff[(long)p * 256 + (c - 3)];
  fxf[t] = v;
}

// vote = fxf + offset; vote_xyz -> d_out & ws; vote_feature = [score, vote[3:]]
// fxf stride 260, offset stride 259.
__global__ void k_vote_finish(const float* __restrict__ fxf, const float* __restrict__ off,
                              const float* __restrict__ score,
                              float* __restrict__ out1, float* __restrict__ votexyz,
                              float* __restrict__ vf) {
  int t = blockIdx.x * blockDim.x + threadIdx.x;
  int total = BATCH * 128 * 259;
  if (t >= total) return;
  int p = t / 259, c = t - p * 259;
  float v = fxf[(long)p * 260 + c] + off[t];
  if (c < 3) { out1[p * 3 + c] = v; votexyz[p * 3 + c] = v; }
  else       vf[(long)p * 257 + (c - 2)] = v;
  if (c == 3) vf[(long)p * 257] = score[p];
}

// boxes = prop(+centers for xyz) -> d_out; centers -> d_out
__global__ void k_prop_out(const float* __restrict__ prop, const float* __restrict__ centers,
                           float* __restrict__ out2, float* __restrict__ out3) {
  int t = blockIdx.x * blockDim.x + threadIdx.x;
  int total = BATCH * 64 * 5;
  if (t >= total) return;
  int p = t / 5, c = t - p * 5;
  out2[t] = prop[t] + (c < 3 ? centers[p * 3 + c] : 0.f);
  if (c < 3) out3[p * 3 + c] = centers[p * 3 + c];
}

// ---------------------------------------------------------------------------
struct Weights3 { const float *w0, *b0, *w1, *b1, *w2, *b2; };

static inline int cdiv(long a, long b) { return (int)((a + b - 1) / b); }

static void gemm(hipStream_t st, const float* X, const float* W, const float* bias,
                 float* Y, int M, int N, int K, int relu,
                 long sX = 0, long sW = 0, long sY = 0, int nb = 1) {
  int groups = (M / 16) * (N / 64);
  dim3 g(cdiv(groups, 8), nb);
  k_gemm_fast<<<g, 256, 0, st>>>(X, W, bias, Y, M, N, K, sX, sW, sY, relu);
}

static void gemm_edge(hipStream_t st, const float* X, const float* W, const float* bias,
                      float* Y, int M, int N, int K) {
  int tiles = (M / 16) * cdiv(N, 16);
  k_gemm_edge<<<cdiv(tiles, 8), 256, 0, st>>>(X, W, bias, Y, M, N, K);
}

static void run_sa(hipStream_t st, const float* xyz, const float* feat,
                   int N, int Cf, int S, float radius, int ns,
                   const float* w0p, Weights3 w, int c1, int c2, int c3,
                   int* fpsIdx, float* newxyz, int* bq, float* outFeat) {
  k_fps<<<BATCH, 256, 0, st>>>(xyz, fpsIdx, N, S);
  k_gather3<<<cdiv(BATCH * S * 3, 256), 256, 0, st>>>(xyz, fpsIdx, newxyz, N, S);
  k_ballq<<<cdiv(BATCH * S, 256), 256, 0, st>>>(xyz, newxyz, bq, N, S, ns, radius * radius);
  int c0 = 3 + Cf, c0p = (c0 + 3) & ~3;
  int maxd = c0p; if (c1 > maxd) maxd = c1; if (c2 > maxd) maxd = c2; if (c3 > maxd) maxd = c3;
  int stride = maxd + 4;                 // even -> b64-aligned, dodges bank camping
  int bufElems = ns * stride;
  size_t ldsB = (size_t)2 * bufElems * sizeof(float);
  dim3 g(S, BATCH);
  k_rows_mlp_max<<<g, 256, ldsB, st>>>(0, xyz, feat, newxyz, bq, nullptr,
                                       S, N, Cf, ns, c0p, c1, c2, c3,
                                       w0p, w.b0, w.w1, w.b1, w.w2, w.b2,
                                       stride, bufElems, outFeat);
}

// ---------------------------------------------------------------------------
extern "C" void kernel_launch(void* const* d_in, const int* in_sizes, int n_in,
                              void* d_out, int out_size, void* d_ws, size_t ws_size,
                              hipStream_t stream) {
  const float* tpl = (const float*)d_in[0];   // (32,512,6)
  const float* sch = (const float*)d_in[1];   // (32,1024,6)
  auto P = [&](int i) { return (const float*)d_in[i]; };
  // params order: sa1(2..7) sa2(8..13) sa3(14..19) cov(20,21) xmlp(22..27)
  //               cla(28..33) fea(34..37) vote(38..43) agg(44..49) prop(50..55)
  Weights3 sa1{P(2),P(3),P(4),P(5),P(6),P(7)};
  Weights3 sa2{P(8),P(9),P(10),P(11),P(12),P(13)};
  Weights3 sa3{P(14),P(15),P(16),P(17),P(18),P(19)};
  const float *covW = P(20), *covB = P(21);
  Weights3 xml{P(22),P(23),P(24),P(25),P(26),P(27)};
  Weights3 cla{P(28),P(29),P(30),P(31),P(32),P(33)};
  const float *feaW0 = P(34), *feaB0 = P(35), *feaW1 = P(36), *feaB1 = P(37);
  Weights3 vot{P(38),P(39),P(40),P(41),P(42),P(43)};
  Weights3 agg{P(44),P(45),P(46),P(47),P(48),P(49)};
  Weights3 prp{P(50),P(51),P(52),P(53),P(54),P(55)};
  (void)in_sizes; (void)n_in; (void)out_size; (void)ws_size;

  // bump arena over d_ws (every buffer fully written before read)
  char* base = (char*)d_ws;
  size_t off = 0;
  auto allocF = [&](size_t n) -> float* {
    float* r = (float*)(base + off);
    off = (off + n * sizeof(float) + 255) & ~(size_t)255;
    return r;
  };
  auto allocI = [&](size_t n) -> int* {
    int* r = (int*)(base + off);
    off = (off + n * sizeof(int) + 255) & ~(size_t)255;
    return r;
  };

  // ---------------- K-padded weight copies (kills all K-loop guards) --------
  float* sa1w0p = allocF(64 * 8);
  float* sa2w0p = allocF(128 * 132);
  float* sa3w0p = allocF(256 * 260);
  float* votw0p = allocF(256 * 260);
  k_padw<<<cdiv(64 * 8, 256), 256, 0, stream>>>(sa1.w0, sa1w0p, 64, 6, 8);
  k_padw<<<cdiv(128 * 132, 256), 256, 0, stream>>>(sa2.w0, sa2w0p, 128, 131, 132);
  k_padw<<<cdiv(256 * 260, 256), 256, 0, stream>>>(sa3.w0, sa3w0p, 256, 259, 260);
  k_padw<<<cdiv(256 * 260, 256), 256, 0, stream>>>(vot.w0, votw0p, 256, 259, 260);

  // ---------------- template backbone ----------------
  float* t_xyz  = allocF(32 * 512 * 3);
  float* t_ft0  = allocF(32 * 512 * 3);
  k_split<<<cdiv(32 * 512 * 3, 256), 256, 0, stream>>>(tpl, t_xyz, t_ft0, 512);
  int*   t_fi   = allocI(32 * 512);
  float* t_new1 = allocF(32 * 256 * 3);
  int*   t_bq1  = allocI(32 * 256 * 32);
  float* t_f1   = allocF(32 * 256 * 128);
  run_sa(stream, t_xyz, t_ft0, 512, 3, 256, 0.3f, 32, sa1w0p, sa1, 64, 64, 128,
         t_fi, t_new1, t_bq1, t_f1);
  float* t_new2 = allocF(32 * 128 * 3);
  int*   t_bq2  = allocI(32 * 128 * 32);
  float* t_f2   = allocF(32 * 128 * 256);
  run_sa(stream, t_new1, t_f1, 256, 128, 128, 0.5f, 32, sa2w0p, sa2, 128, 128, 256,
         t_fi, t_new2, t_bq2, t_f2);
  float* t_new3 = allocF(32 * 64 * 3);     // template_xyz
  int*   t_bq3  = allocI(32 * 64 * 32);
  float* t_f3   = allocF(32 * 64 * 256);
  run_sa(stream, t_new2, t_f2, 128, 256, 64, 0.7f, 32, sa3w0p, sa3, 256, 256, 256,
         t_fi, t_new3, t_bq3, t_f3);
  float* t_fin  = allocF(32 * 64 * 256);   // template_feature (B,64,256)
  gemm(stream, t_f3, covW, covB, t_fin, 32 * 64, 256, 256, 0);

  // ---------------- search backbone ----------------
  float* s_xyz  = allocF(32 * 1024 * 3);
  float* s_ft0  = allocF(32 * 1024 * 3);
  k_split<<<cdiv(32 * 1024 * 3, 256), 256, 0, stream>>>(sch, s_xyz, s_ft0, 1024);
  int*   s_fi   = allocI(32 * 1024);
  float* s_new1 = allocF(32 * 512 * 3);
  int*   s_bq1  = allocI(32 * 512 * 32);
  float* s_f1   = allocF(32 * 512 * 128);
  run_sa(stream, s_xyz, s_ft0, 1024, 3, 512, 0.3f, 32, sa1w0p, sa1, 64, 64, 128,
         s_fi, s_new1, s_bq1, s_f1);
  float* s_new2 = allocF(32 * 256 * 3);
  int*   s_bq2  = allocI(32 * 256 * 32);
  float* s_f2   = allocF(32 * 256 * 256);
  run_sa(stream, s_new1, s_f1, 512, 128, 256, 0.5f, 32, sa2w0p, sa2, 128, 128, 256,
         s_fi, s_new2, s_bq2, s_f2);
  float* s_new3 = allocF(32 * 128 * 3);    // search_xyz
  int*   s_bq3  = allocI(32 * 128 * 32);
  float* s_f3   = allocF(32 * 128 * 256);
  run_sa(stream, s_new2, s_f2, 256, 256, 128, 0.7f, 32, sa3w0p, sa3, 256, 256, 256,
         s_fi, s_new3, s_bq3, s_f3);
  float* s_fin  = allocF(32 * 128 * 256);  // search_feature (B,128,256)
  gemm(stream, s_f3, covW, covB, s_fin, 32 * 128, 256, 256, 0);

  // ---------------- xcorr ----------------
  float* onb = allocF(32 * 64 * 256);
  float* lnb = allocF(32 * 128 * 256);
  k_norm<<<cdiv(32 * 64, 256), 256, 0, stream>>>(t_fin, onb, 32 * 64, 256);
  k_norm<<<cdiv(32 * 128, 256), 256, 0, stream>>>(s_fin, lnb, 32 * 128, 256);
  float* cosb = allocF(32 * 64 * 128);
  // cos[b] (64x128) = on[b] (64x256) x ln[b]^T, batched over 32
  gemm(stream, onb, lnb, nullptr, cosb, 64, 128, 256, 0,
       (long)64 * 256, (long)128 * 256, (long)64 * 128, 32);
  float* fmaxb = allocF(32 * 128 * 256);   // max over n1 of xmlp output
  {
    int stride = 264, bufElems = 64 * stride;
    dim3 g(128, BATCH);
    k_rows_mlp_max<<<g, 256, (size_t)2 * bufElems * sizeof(float), stream>>>(
        1, nullptr, t_fin, t_new3, nullptr, cosb,
        128, 64, 256, 64, 260, 256, 256, 256,
        xml.w0, xml.b0, xml.w1, xml.b1, xml.w2, xml.b2,
        stride, bufElems, fmaxb);
  }
  float* tmpA = allocF(32 * 128 * 256);
  float* tmpB = allocF(32 * 128 * 256);
  float* ff   = allocF(32 * 128 * 256);    // fusion_feature
  gemm(stream, fmaxb, feaW0, feaB0, tmpA, 4096, 256, 256, 1);
  gemm(stream, tmpA, feaW1, feaB1, ff, 4096, 256, 256, 0);

  // ---------------- cla head ----------------
  gemm(stream, ff, cla.w0, cla.b0, tmpA, 4096, 256, 256, 1);
  gemm(stream, tmpA, cla.w1, cla.b1, tmpB, 4096, 256, 256, 1);
  float* claout = allocF(4096);
  gemm_edge(stream, tmpB, cla.w2, cla.b2, claout, 4096, 1, 256);
  float* score = allocF(4096);
  k_cla_sig<<<cdiv(4096, 256), 256, 0, stream>>>(claout, (float*)d_out, score);

  // ---------------- vote head ----------------
  float* fxf = allocF((size_t)4096 * 260);
  k_build_fxf<<<cdiv(4096 * 260, 256), 256, 0, stream>>>(s_new3, ff, fxf);
  gemm(stream, fxf, votw0p, vot.b0, tmpA, 4096, 256, 260, 1);   // padded K
  gemm(stream, tmpA, vot.w1, vot.b1, tmpB, 4096, 256, 256, 1);
  float* voff = allocF((size_t)4096 * 259);
  gemm_edge(stream, tmpB, vot.w2, vot.b2, voff, 4096, 259, 256);
  float* votexyz = allocF(4096 * 3);
  float* vfeat   = allocF((size_t)4096 * 257);
  k_vote_finish<<<cdiv(4096 * 259, 256), 256, 0, stream>>>(
      fxf, voff, score, (float*)d_out + 4096, votexyz, vfeat);

  // ---------------- aggregation SA ----------------
  int*   a_fi    = allocI(32 * 64);
  float* centers = allocF(32 * 64 * 3);
  int*   a_bq    = allocI(32 * 64 * 16);
  float* pf      = allocF(32 * 64 * 256);
  run_sa(stream, votexyz, vfeat, 128, 257, 64, 0.3f, 16, agg.w0, agg, 256, 256, 256,
         a_fi, centers, a_bq, pf);

  // ---------------- proposal head ----------------
  gemm(stream, pf, prp.w0, prp.b0, tmpA, 2048, 256, 256, 1);
  gemm(stream, tmpA, prp.w1, prp.b1, tmpB, 2048, 256, 256, 1);
  float* prop = allocF(2048 * 5);
  gemm_edge(stream, tmpB, prp.w2, prp.b2, prop, 2048, 5, 256);
  k_prop_out<<<cdiv(2048 * 5, 256), 256, 0, stream>>>(
      prop, centers, (float*)d_out + 16384, (float*)d_out + 26624);
}